// preAttention_13477607375310
// MI455X (gfx1250) — compile-verified
//
#include <hip/hip_runtime.h>
#include <math.h>

// ---------------- problem constants ----------------
#define WS      8
#define NHEADS  16
#define CIN     256
#define COUT    512
#define HDIM    16          // head dim d
#define NTOK    64          // ws*ws tokens per window
#define HH      24          // 192/8
#define WWIN    24
#define BATCH   2
#define NWIN    (BATCH*HH*WWIN)     // 1152
#define HIMG    192
#define WIMG    192
#define BN_EPS  1e-5f

// ---------------- LDS layout (floats) ----------------
// x staged by TDM, channel-major with HW padding: x[c*XSTRIDE + t], XSTRIDE = 64+8
// (pad_interval=64 DW, pad_amount=8 DW -> A-frag lane-group bank sets disjoint)
#define XSTRIDE   72
#define XREGION   (CIN*XSTRIDE)                 // 18432 floats (aliases q region + slack)
#define QKSTRIDE  17                            // q/k token row stride (gcd(17,64)=1)
#define QOFF      0                             // q: 16*64*17 = 17408 floats (inside x region)
#define KOFF      XREGION                       // 18432
#define AVGOFF    (KOFF + NHEADS*NTOK*QKSTRIDE) // 35840
#define RELSZ     ((2*WS-1)*(2*WS-1)*NHEADS)    // 225*16 = 3600
#define RELOFF    (AVGOFF + NTOK)               // 35904
#define SMEM_FLOATS (RELOFF + RELSZ)            // 39504 floats = 158,016 B (< 320 KB/WGP)

typedef __attribute__((ext_vector_type(2))) float v2f;
typedef __attribute__((ext_vector_type(8))) float v8f;
typedef __attribute__((ext_vector_type(4))) unsigned int u32x4;
typedef __attribute__((ext_vector_type(4))) int i32x4;
typedef __attribute__((ext_vector_type(8))) int i32x8;

__global__ __launch_bounds__(256, 1)
void fused_window_preattn(const float* __restrict__ x,
                          const float* __restrict__ w_qk,
                          const float* __restrict__ gamma,
                          const float* __restrict__ beta,
                          const float* __restrict__ rmean,
                          const float* __restrict__ rvar,
                          const float* __restrict__ rel_table,
                          float* __restrict__ out_attn,
                          float* __restrict__ out_avg)
{
    extern __shared__ float smem[];
    const int tid  = threadIdx.x;
    const int lane = tid & 31;
    const int wave = tid >> 5;          // 0..7
    const int lg   = lane >> 4;         // 0/1 : WMMA lane half
    const int ll   = lane & 15;

    const int win = blockIdx.x;
    const int b   = win / (HH * WWIN);
    const int rem = win % (HH * WWIN);
    const int wy  = rem / WWIN;
    const int wx  = rem % WWIN;

    // =========== Stage 1: TDM async tensor load of x window -> LDS ===========
    // 3D tile: X=8 (cols), Y=8 (rows, stride 192), Z=256 (channels, stride 192*192),
    // LDS dest padded 8 DW per 64 DW -> x_lds[c*72 + (y*8+x)]
    const float* xb = x + (size_t)b * CIN * HIMG * WIMG;
    if (wave == 0) {
        unsigned long long ga =
            (unsigned long long)(uintptr_t)(xb + (size_t)(wy * WS) * WIMG + wx * WS);
        unsigned lds_off = (unsigned)(uintptr_t)(void*)smem;   // LDS byte offset (x at base)

        u32x4 g0;
        g0[0] = 1u;                                            // count=1 valid user D#
        g0[1] = lds_off;                                       // lds_addr (bytes)
        g0[2] = (unsigned)(ga & 0xffffffffull);                // global_addr[31:0]
        g0[3] = (unsigned)((ga >> 32) & 0x1ffffffull)          // global_addr[56:32]
              | (2u << 30);                                    // type=2 (image)

        i32x8 g1;
        g1[0] = (int)((2u << 16)      // data_size = 4B
                    | (1u << 20)      // pad_enable
                    | (5u << 22)      // pad_interval: 2^(5+1)=64 DW
                    | (7u << 25));    // pad_amount: 7 -> 8 DW
        g1[1] = (int)(((unsigned)WIMG & 0xffffu) << 16);       // tensor_dim0 lo16 (192)
        g1[2] = (int)(((unsigned)HIMG & 0xffffu) << 16);       // dim0 hi16=0 | tensor_dim1 lo16 (192)
        g1[3] = (int)(8u << 16);                               // dim1 hi16=0 | tile_dim0=8
        g1[4] = (int)(8u | (256u << 16));                      // tile_dim1=8 | tile_dim2=256
        g1[5] = (int)WIMG;                                     // tensor_dim0_stride lo32 = 192
        g1[6] = (int)(((unsigned)(HIMG * WIMG) & 0xffffu) << 16); // d0s hi16=0 | d1s lo16 (36864)
        g1[7] = (int)(((unsigned)(HIMG * WIMG)) >> 16);        // tensor_dim1_stride[47:16]

        i32x4 g2;
        g2[0] = CIN;                                           // tensor_dim2 = 256
        g2[1] = 1;                                             // tensor_dim3 (unused)
        g2[2] = 0;                                             // tensor_dim2_stride lo (unused)
        g2[3] = 0;                                             // d2s hi | tile_dim3=0
        i32x4 g3; g3[0] = 0; g3[1] = 0; g3[2] = 0; g3[3] = 0;

#if defined(__clang_major__) && (__clang_major__ >= 23)
        i32x8 gz; for (int i = 0; i < 8; ++i) gz[i] = 0;
        __builtin_amdgcn_tensor_load_to_lds(g0, g1, g2, g3, gz, 0);
#else
        __builtin_amdgcn_tensor_load_to_lds(g0, g1, g2, g3, 0);
#endif
    }

    // overlap with TDM: stage rel_table, zero avg accumulator
    for (int i = tid; i < RELSZ; i += 256) smem[RELOFF + i] = rel_table[i];
    if (tid < NTOK) smem[AVGOFF + tid] = 0.0f;

    if (wave == 0) __builtin_amdgcn_s_wait_tensorcnt(0);
    __syncthreads();

    // =========== Stage 2: 1x1 conv GEMM (64 tok x 64 out-ch per wave, K=256) ===========
    const int obase = wave * 64;        // waves 0-3 -> q channels, 4-7 -> k channels
    v8f acc[4][4];
    {
        v8f vz;
        #pragma unroll
        for (int i = 0; i < 8; ++i) vz[i] = 0.0f;
        #pragma unroll
        for (int mt = 0; mt < 4; ++mt)
            #pragma unroll
            for (int nt = 0; nt < 4; ++nt) acc[mt][nt] = vz;
    }

    for (int k0 = 0; k0 < CIN; k0 += 4) {
        const int kb = k0 + 2 * lg;
        v2f afr[4], bfr[4];
        #pragma unroll
        for (int mt = 0; mt < 4; ++mt) {                 // A = x window (tok x ch), channel-major LDS
            const float* p = &smem[(size_t)kb * XSTRIDE + (mt * 16 + ll)];
            afr[mt].x = p[0];
            afr[mt].y = p[XSTRIDE];
        }
        #pragma unroll
        for (int nt = 0; nt < 4; ++nt) {                 // B = w_qk^T (ch_in x ch_out), L2-resident
            // contiguous 8B per lane, kb even -> single global_load_b64
            bfr[nt] = *(const v2f*)&w_qk[(size_t)(obase + nt * 16 + ll) * CIN + kb];
        }
        #pragma unroll
        for (int mt = 0; mt < 4; ++mt)
            #pragma unroll
            for (int nt = 0; nt < 4; ++nt)
                acc[mt][nt] = __builtin_amdgcn_wmma_f32_16x16x4_f32(
                    false, afr[mt], false, bfr[nt], (short)0, acc[mt][nt], false, false);
    }

    // ---- BN(eval) + ReLU in registers (per-lane channel o = obase + nt*16 + ll) ----
    #pragma unroll
    for (int nt = 0; nt < 4; ++nt) {
        const int o   = obase + nt * 16 + ll;
        const float inv = gamma[o] * rsqrtf(rvar[o] + BN_EPS);
        const float add = beta[o] - rmean[o] * inv;
        #pragma unroll
        for (int mt = 0; mt < 4; ++mt)
            #pragma unroll
            for (int r = 0; r < 8; ++r) {
                float v = acc[mt][nt][r] * inv + add;
                acc[mt][nt][r] = v > 0.0f ? v : 0.0f;
            }
    }
    __syncthreads();   // all waves finished reading x_lds

    // ---- scatter q/k to LDS: [h][token*17 + d] (q aliases dead x region) ----
    #pragma unroll
    for (int nt = 0; nt < 4; ++nt) {
        const int o    = obase + nt * 16 + ll;
        const int base = (o < CIN) ? QOFF : KOFF;
        const int oc   = o & (CIN - 1);
        const int h    = oc >> 4, d = oc & 15;
        #pragma unroll
        for (int mt = 0; mt < 4; ++mt)
            #pragma unroll
            for (int r = 0; r < 8; ++r) {
                const int t = mt * 16 + r + 8 * lg;
                smem[base + h * (NTOK * QKSTRIDE) + t * QKSTRIDE + d] = acc[mt][nt][r];
            }
    }
    __syncthreads();

    // =========== Stage 3: per-head dots + bias + avg + softmax ===========
    const float scale = 0.25f;                       // d^-0.5, d=16
    float* attn_base = out_attn + (size_t)win * NHEADS * NTOK * NTOK;

    for (int hi = 0; hi < 2; ++hi) {
        const int h = wave * 2 + hi;
        v8f dacc[4][4];
        {
            v8f vz;
            #pragma unroll
            for (int i = 0; i < 8; ++i) vz[i] = 0.0f;
            #pragma unroll
            for (int mt = 0; mt < 4; ++mt)
                #pragma unroll
                for (int nt = 0; nt < 4; ++nt) dacc[mt][nt] = vz;
        }

        #pragma unroll
        for (int k0 = 0; k0 < HDIM; k0 += 4) {
            const int kb = k0 + 2 * lg;
            v2f afr[4], bfr[4];
            #pragma unroll
            for (int mt = 0; mt < 4; ++mt) {          // A = q[h] (query x d)
                const float* p = &smem[QOFF + h * (NTOK * QKSTRIDE) + (mt * 16 + ll) * QKSTRIDE + kb];
                afr[mt].x = p[0]; afr[mt].y = p[1];
            }
            #pragma unroll
            for (int nt = 0; nt < 4; ++nt) {          // B = k[h]^T (d x key)
                const float* p = &smem[KOFF + h * (NTOK * QKSTRIDE) + (nt * 16 + ll) * QKSTRIDE + kb];
                bfr[nt].x = p[0]; bfr[nt].y = p[1];
            }
            #pragma unroll
            for (int mt = 0; mt < 4; ++mt)
                #pragma unroll
                for (int nt = 0; nt < 4; ++nt)
                    dacc[mt][nt] = __builtin_amdgcn_wmma_f32_16x16x4_f32(
                        false, afr[mt], false, bfr[nt], (short)0, dacc[mt][nt], false, false);
        }

        // ---- scale + relative-position bias ----
        #pragma unroll
        for (int nt = 0; nt < 4; ++nt) {
            const int kk = nt * 16 + ll;              // key token
            const int ik = kk >> 3, jk = kk & 7;
            #pragma unroll
            for (int mt = 0; mt < 4; ++mt)
                #pragma unroll
                for (int r = 0; r < 8; ++r) {
                    const int q  = mt * 16 + r + 8 * lg;   // query token
                    const int iq = q >> 3, jq = q & 7;
                    const int idx = (iq - ik + WS - 1) * (2 * WS - 1) + (jq - jk + WS - 1);
                    dacc[mt][nt][r] = dacc[mt][nt][r] * scale + smem[RELOFF + idx * NHEADS + h];
                }
        }

        // ---- avg accumulation (sum over heads & queries per key) ----
        #pragma unroll
        for (int nt = 0; nt < 4; ++nt) {
            float part = 0.0f;
            #pragma unroll
            for (int mt = 0; mt < 4; ++mt)
                #pragma unroll
                for (int r = 0; r < 8; ++r) part += dacc[mt][nt][r];
            atomicAdd(&smem[AVGOFF + nt * 16 + ll], part);   // ds_add_f32
        }

        // ---- row softmax over 64 keys (4 tiles in-reg + 16-lane shfl_xor) ----
        #pragma unroll
        for (int mt = 0; mt < 4; ++mt)
            #pragma unroll
            for (int r = 0; r < 8; ++r) {
                const int q = mt * 16 + r + 8 * lg;
                float v0 = dacc[mt][0][r], v1 = dacc[mt][1][r];
                float v2 = dacc[mt][2][r], v3 = dacc[mt][3][r];
                float m = fmaxf(fmaxf(v0, v1), fmaxf(v2, v3));
                #pragma unroll
                for (int off = 1; off < 16; off <<= 1) m = fmaxf(m, __shfl_xor(m, off, 32));
                const float e0 = __expf(v0 - m), e1 = __expf(v1 - m);
                const float e2 = __expf(v2 - m), e3 = __expf(v3 - m);
                float s = (e0 + e1) + (e2 + e3);
                #pragma unroll
                for (int off = 1; off < 16; off <<= 1) s += __shfl_xor(s, off, 32);
                const float rs = 1.0f / s;
                float* op = attn_base + ((size_t)h * NTOK + q) * NTOK;
                // 302 MB one-shot stream: bypass-friendly non-temporal stores
                __builtin_nontemporal_store(e0 * rs, op + 0 * 16 + ll);
                __builtin_nontemporal_store(e1 * rs, op + 1 * 16 + ll);
                __builtin_nontemporal_store(e2 * rs, op + 2 * 16 + ll);
                __builtin_nontemporal_store(e3 * rs, op + 3 * 16 + ll);
            }
    }

    __syncthreads();
    if (tid < NTOK)
        out_avg[(size_t)win * NTOK + tid] = smem[AVGOFF + tid] * (1.0f / (NHEADS * NTOK));
}

extern "C" void kernel_launch(void* const* d_in, const int* in_sizes, int n_in,
                              void* d_out, int out_size, void* d_ws, size_t ws_size,
                              hipStream_t stream) {
    (void)in_sizes; (void)n_in; (void)out_size; (void)d_ws; (void)ws_size;
    const float* x     = (const float*)d_in[0];
    const float* w_qk  = (const float*)d_in[1];
    const float* gam   = (const float*)d_in[2];
    const float* bet   = (const float*)d_in[3];
    const float* rmean = (const float*)d_in[4];
    const float* rvar  = (const float*)d_in[5];
    const float* rel   = (const float*)d_in[6];
    float* out_attn = (float*)d_out;
    float* out_avg  = out_attn + (size_t)NWIN * NHEADS * NTOK * NTOK;  // concat order: attn, avg

    const size_t smem_bytes = SMEM_FLOATS * sizeof(float);
    fused_window_preattn<<<NWIN, 256, smem_bytes, stream>>>(
        x, w_qk, gam, bet, rmean, rvar, rel, out_attn, out_avg);
}